// LinearCRF_73400991089070
// MI455X (gfx1250) — compile-verified
//
#include <hip/hip_runtime.h>

#define B_SZ 64
#define T_SZ 512
#define L_SZ 48
#define START_I 45
#define END_I 46

typedef float v2f __attribute__((ext_vector_type(2)));
typedef float v8f __attribute__((ext_vector_type(8)));

// ---------------------------------------------------------------------------
// Partition function (unlabeled score) via WMMA-accelerated forward scan.
// Grid: 4 blocks x 96 threads (3 full waves). Block b handles batches
// [16b, 16b+16). Each wave owns one 16-wide N-tile of the 48 labels.
// Per time step: S = P * expT  (P[b,i] = exp(alpha[b,i]-m_b), expT = exp(trans))
// alpha_new[b,j] = emit[b,t,j] + m_b + log(S[b,j]).
// Emit tiles are software-pipelined one step ahead (global latency overlaps
// the WMMA + transcendental work of the current step).
// ---------------------------------------------------------------------------
__global__ __launch_bounds__(96) void crf_forward_wmma_kernel(
    const float* __restrict__ lstm, const float* __restrict__ trans,
    const int* __restrict__ lens, float* __restrict__ partial)
{
    __shared__ float expT[L_SZ][L_SZ];   // exp(transition)
    __shared__ float alpha[16][L_SZ];
    __shared__ float Pbuf[16][L_SZ];
    __shared__ float emitT[16][L_SZ];
    __shared__ float finalA[16][L_SZ];   // alpha row at t = len-1
    __shared__ float pmax[6][16];        // partial row maxima
    __shared__ float mrow[16];
    __shared__ int   slen[16];
    __shared__ float red[16];

    const int tid  = threadIdx.x;
    const int base = blockIdx.x * 16;

    // --- one-time setup: expT, seq lens, alpha at t=0 ---
    for (int idx = tid; idx < L_SZ * L_SZ; idx += 96)
        expT[idx / L_SZ][idx % L_SZ] = __expf(trans[idx]);
    if (tid < 16) slen[tid] = lens[base + tid];
    for (int idx = tid; idx < 16 * L_SZ; idx += 96) {
        int r = idx / L_SZ, c = idx % L_SZ;
        alpha[r][c] = trans[START_I * L_SZ + c] +
                      lstm[(base + r) * (T_SZ * L_SZ) + c];
    }
    __syncthreads();
    for (int idx = tid; idx < 16 * L_SZ; idx += 96) {
        int r = idx / L_SZ, c = idx % L_SZ;
        if (slen[r] == 1) finalA[r][c] = alpha[r][c];
    }

    const int lane  = tid & 31;
    const int wave  = tid >> 5;            // 0..2 -> N tile
    const int rowA  = lane & 15;           // A-matrix row (batch within tile)
    const int kHalf = (lane >> 4) * 2;     // K pair selected by lane half
    const int colB  = wave * 16 + (lane & 15);
    const int rHalf = (lane >> 4) * 8;     // D-matrix row offset by lane half
    const int grp   = tid >> 4;            // 0..5, max-reduction group
    const int grow  = tid & 15;            // row for max reduction

    // preload emit tile for t = 1
    float eR[8];
    for (int k = 0; k < 8; ++k) {
        int idx = tid + 96 * k;
        int r = idx / L_SZ, c = idx % L_SZ;
        eR[k] = lstm[(base + r) * (T_SZ * L_SZ) + L_SZ + c];
    }
    __syncthreads();

    for (int t = 1; t < T_SZ; ++t) {
        // phase A1: parallel row max (6 partials of 8 elements per row)
        {
            float pm = -3.402823466e38f;
            for (int c = grp * 8; c < grp * 8 + 8; ++c)
                pm = fmaxf(pm, alpha[grow][c]);
            pmax[grp][grow] = pm;
        }
        __syncthreads();
        if (tid < 16) {
            float m = pmax[0][tid];
            for (int g = 1; g < 6; ++g) m = fmaxf(m, pmax[g][tid]);
            mrow[tid] = m;
        }
        __syncthreads();
        // phase A2: P = exp(alpha - m); commit prefetched emit tile to LDS
        for (int k = 0; k < 8; ++k) {
            int idx = tid + 96 * k;
            int r = idx / L_SZ, c = idx % L_SZ;
            Pbuf[r][c]  = __expf(alpha[r][c] - mrow[r]);
            emitT[r][c] = eR[k];
        }
        __syncthreads();

        // issue next step's emit loads; latency hides under WMMA + log below
        if (t + 1 < T_SZ) {
            for (int k = 0; k < 8; ++k) {
                int idx = tid + 96 * k;
                int r = idx / L_SZ, c = idx % L_SZ;
                eR[k] = lstm[(base + r) * (T_SZ * L_SZ) + (t + 1) * L_SZ + c];
            }
        }

        // phase B: S(16x16 per wave) = P(16x48) * expT(48 x 16-tile), f32 WMMA
        v8f acc = {0.f, 0.f, 0.f, 0.f, 0.f, 0.f, 0.f, 0.f};
        for (int kc = 0; kc < 12; ++kc) {
            int kb = kc * 4 + kHalf;
            v2f a;  a.x  = Pbuf[rowA][kb];   a.y  = Pbuf[rowA][kb + 1];
            v2f bm; bm.x = expT[kb][colB];   bm.y = expT[kb + 1][colB];
            acc = __builtin_amdgcn_wmma_f32_16x16x4_f32(
                false, a, false, bm, (short)0, acc, false, false);
        }

        // phase C: alpha_new = emit + m + log(S); fused last-step snapshot
        for (int r = 0; r < 8; ++r) {
            int row = r + rHalf;
            float v = emitT[row][colB] + mrow[row] + __logf(acc[r]);
            alpha[row][colB] = v;
            if (slen[row] == t + 1) finalA[row][colB] = v;
        }
        __syncthreads();
    }

    // per-batch logsumexp_j(finalA + trans[:,END]); deterministic block sum
    if (tid < 16) {
        float m = -3.402823466e38f;
        for (int j = 0; j < L_SZ; ++j)
            m = fmaxf(m, finalA[tid][j] + trans[j * L_SZ + END_I]);
        float s = 0.f;
        for (int j = 0; j < L_SZ; ++j)
            s += __expf(finalA[tid][j] + trans[j * L_SZ + END_I] - m);
        red[tid] = m + __logf(s);
    }
    __syncthreads();
    if (tid == 0) {
        float s = 0.f;
        for (int i = 0; i < 16; ++i) s += red[i];
        partial[blockIdx.x] = s;
    }
}

// ---------------------------------------------------------------------------
// Gold-path (labeled) score: gathers + masked sum. Single block, deterministic.
// ---------------------------------------------------------------------------
__global__ __launch_bounds__(256) void crf_labeled_kernel(
    const float* __restrict__ lstm, const float* __restrict__ trans,
    const int* __restrict__ lens, const int* __restrict__ tags,
    float* __restrict__ out)
{
    const int tid = threadIdx.x;
    float acc = 0.f;

    for (int b = tid; b < B_SZ; b += 256) {
        int len = lens[b];
        int t0  = tags[b * T_SZ];
        acc += trans[START_I * L_SZ + t0] + lstm[b * (T_SZ * L_SZ) + t0];
        int te = tags[b * T_SZ + len - 1];
        acc += trans[te * L_SZ + END_I];
    }
    const int total = B_SZ * (T_SZ - 1);
    for (int i = tid; i < total; i += 256) {
        int b = i / (T_SZ - 1);
        int t = 1 + i % (T_SZ - 1);
        if (t < lens[b]) {
            int prev = tags[b * T_SZ + t - 1];
            int cur  = tags[b * T_SZ + t];
            acc += trans[prev * L_SZ + cur] +
                   lstm[(b * T_SZ + t) * L_SZ + cur];
        }
    }

    __shared__ float red[256];
    red[tid] = acc;
    __syncthreads();
    for (int s = 128; s > 0; s >>= 1) {
        if (tid < s) red[tid] += red[tid + s];
        __syncthreads();
    }
    if (tid == 0) out[1] = red[0];
}

// ---------------------------------------------------------------------------
// Combine the 4 per-block partition partials in fixed order -> out[0].
// ---------------------------------------------------------------------------
__global__ void crf_combine_kernel(const float* __restrict__ ws,
                                   float* __restrict__ out)
{
    if (threadIdx.x == 0)
        out[0] = ((ws[0] + ws[1]) + ws[2]) + ws[3];
}

extern "C" void kernel_launch(void* const* d_in, const int* in_sizes, int n_in,
                              void* d_out, int out_size, void* d_ws, size_t ws_size,
                              hipStream_t stream)
{
    (void)in_sizes; (void)n_in; (void)out_size; (void)ws_size;
    const float* lstm  = (const float*)d_in[0];
    const float* trans = (const float*)d_in[1];
    const int*   lens  = (const int*)d_in[2];
    const int*   tags  = (const int*)d_in[3];
    // d_in[4] (mask) is derived from lens; recomputed on the fly.
    float* out = (float*)d_out;
    float* ws  = (float*)d_ws;

    crf_forward_wmma_kernel<<<4, 96, 0, stream>>>(lstm, trans, lens, ws);
    crf_labeled_kernel<<<1, 256, 0, stream>>>(lstm, trans, lens, tags, out);
    crf_combine_kernel<<<1, 32, 0, stream>>>(ws, out);
}